// VFR_83803401880152
// MI455X (gfx1250) — compile-verified
//
#include <hip/hip_runtime.h>

typedef float v2f __attribute__((ext_vector_type(2)));
typedef float v8f __attribute__((ext_vector_type(8)));

#define BDIM 8
#define NPTS 8192
#define KNN  16
#define CH   64
#define MTOK (BDIM * NPTS)   // 65536 tokens
#define ROWP 66              // padded LDS row stride (floats) to avoid bank conflicts
#define BN_EPS 1e-5f

// ---------------- Kernel 0: zero the per-channel sum / sumsq accumulators ----
__global__ void k0_zero(float* __restrict__ stats) {
    int i = threadIdx.x;
    if (i < 2 * CH) stats[i] = 0.0f;
}

// ---------------- Kernel 1: gather-mean + WMMA GEMM + stat accumulation ------
// One wave (32 lanes) handles one 16-token tile. Block = 4 waves.
__global__ __launch_bounds__(128) void k1_gemm(
        const float* __restrict__ x,
        const long long* __restrict__ knn,
        const float* __restrict__ W,
        float* __restrict__ agg,
        float* __restrict__ stats) {
    __shared__ float wt[CH][ROWP];      // W row-major, padded
    __shared__ float xt[4][16][ROWP];   // per-wave xbar tile: 16 tokens x 64 ch

    const int tid  = threadIdx.x;
    const int lane = tid & 31;
    const int wave = tid >> 5;

    // Cooperative load of W (64x64 f32 = 16 KB) into LDS, float2 per thread.
    for (int i = tid; i < (CH * CH) / 2; i += 128) {
        int r = (i * 2) >> 6;
        int c = (i * 2) & 63;
        *(v2f*)&wt[r][c] = *(const v2f*)&W[i * 2];
    }

    const int tile = blockIdx.x * 4 + wave;  // 16-token tile id
    const int m0   = tile * 16;
    const int b    = m0 >> 13;               // NPTS = 8192 = 2^13, 16 | 8192
    const int n0   = m0 & (NPTS - 1);

    // Gather + mean (linearity: average x before the linear layer).
    // Lane L accumulates channels {2L, 2L+1} of each token: one b64 load per
    // neighbor row -> a single coalesced 256B request per row across the wave.
    for (int t = 0; t < 16; ++t) {
        const long long* kp = knn + ((long long)(b * NPTS + n0 + t)) * KNN;
        int myidx = (int)kp[lane & 15];      // coalesced 16x int64 load
        v2f a = {0.f, 0.f};
        #pragma unroll
        for (int k = 0; k < KNN; ++k) {
            int idx = __shfl(myidx, k);
            const float* xr = x + ((long long)(b * NPTS + idx)) * CH;
            a += *(const v2f*)(xr + lane * 2);
        }
        a *= (1.0f / KNN);
        *(v2f*)&xt[wave][t][lane * 2] = a;   // 8B-aligned (row stride 264B)
    }
    __syncthreads();   // make LDS writes (W + cross-lane xbar) visible

    // WMMA f32 16x16x4: A = xbar[16x4 chunk], B = W^T[4x16 chunk].
    // A layout: lanes 0-15 -> M=lane, K = 4s+{0,1}; lanes 16-31 -> K = 4s+{2,3}.
    // B layout: lanes 0-15 -> N=lane, K = 4s+{0,1}; lanes 16-31 -> K = 4s+{2,3}.
    // B[k][n] = W[o = 16j+n][c = k]  (h = xbar * W^T)
    const int mrow = lane & 15;
    const int koff = (lane < 16) ? 0 : 2;

    v8f acc[4];
    const v8f vzero = {0.f, 0.f, 0.f, 0.f, 0.f, 0.f, 0.f, 0.f};
    #pragma unroll
    for (int j = 0; j < 4; ++j) acc[j] = vzero;

    #pragma unroll
    for (int s = 0; s < 16; ++s) {
        v2f a = *(const v2f*)&xt[wave][mrow][4 * s + koff];
        #pragma unroll
        for (int j = 0; j < 4; ++j) {
            v2f bv = *(const v2f*)&wt[j * 16 + mrow][4 * s + koff];
            acc[j] = __builtin_amdgcn_wmma_f32_16x16x4_f32(
                         false, a, false, bv, (short)0, acc[j], false, false);
        }
    }

    // C/D layout: lane L (<16) holds rows 0..7, lane L+16 rows 8..15, same col.
    const int mbase = (lane < 16) ? 0 : 8;
    #pragma unroll
    for (int j = 0; j < 4; ++j) {
        const int o = j * 16 + mrow;         // output channel
        float ps = 0.f, pq = 0.f;
        #pragma unroll
        for (int r = 0; r < 8; ++r) {
            float v = acc[j][r];
            agg[(long long)(m0 + mbase + r) * CH + o] = v;
            ps += v;
            pq += v * v;
        }
        ps += __shfl_xor(ps, 16);            // combine row halves (same channel)
        pq += __shfl_xor(pq, 16);
        if (lane < 16) {
            atomicAdd(&stats[o], ps);
            atomicAdd(&stats[CH + o], pq);
        }
    }
}

// ---------------- Kernel 2: fold stats + gamma/beta into scale/shift ---------
__global__ void k2_bnparams(const float* __restrict__ stats,
                            const float* __restrict__ gamma,
                            const float* __restrict__ beta,
                            float* __restrict__ ss) {
    int c = threadIdx.x;
    if (c < CH) {
        const float inv = 1.0f / (float)MTOK;
        float mu  = stats[c] * inv;
        float var = stats[CH + c] * inv - mu * mu;   // biased variance
        float rs  = rsqrtf(var + BN_EPS);
        float sc  = gamma[c] * rs;
        ss[c]      = sc;
        ss[CH + c] = beta[c] - mu * sc;
    }
}

// ---------------- Kernel 3: normalize (float4 vectorized) --------------------
__global__ __launch_bounds__(256) void k3_norm(const float* __restrict__ agg,
                                               const float* __restrict__ ss,
                                               float* __restrict__ out) {
    long long i4 = ((long long)blockIdx.x * 256 + threadIdx.x) * 4;
    int c = (int)(i4 & (CH - 1));            // 4 | 64 -> float4 never crosses a row
    float4 a  = *(const float4*)(agg + i4);
    float4 sc = *(const float4*)(ss + c);
    float4 sh = *(const float4*)(ss + CH + c);
    float4 o;
    o.x = a.x * sc.x + sh.x;
    o.y = a.y * sc.y + sh.y;
    o.z = a.z * sc.z + sh.z;
    o.w = a.w * sc.w + sh.w;
    *(float4*)(out + i4) = o;
}

// ---------------- Host launcher ----------------------------------------------
extern "C" void kernel_launch(void* const* d_in, const int* in_sizes, int n_in,
                              void* d_out, int out_size, void* d_ws, size_t ws_size,
                              hipStream_t stream) {
    const float*     x     = (const float*)d_in[0];
    const long long* knn   = (const long long*)d_in[1];   // int64 per reference
    const float*     W     = (const float*)d_in[2];
    const float*     gamma = (const float*)d_in[3];
    const float*     beta  = (const float*)d_in[4];
    float* out = (float*)d_out;

    char*  ws    = (char*)d_ws;
    float* agg   = (float*)ws;                                  // 16 MB
    float* stats = (float*)(ws + (size_t)MTOK * CH * sizeof(float));  // 128 f32
    float* ss    = stats + 2 * CH;                              // 128 f32

    k0_zero    <<<1, 128, 0, stream>>>(stats);
    k1_gemm    <<<MTOK / 16 / 4, 128, 0, stream>>>(x, knn, W, agg, stats);
    k2_bnparams<<<1, 64, 0, stream>>>(stats, gamma, beta, ss);
    k3_norm    <<<(MTOK * CH / 4) / 256, 256, 0, stream>>>(agg, ss, out);
}